// FCOSPostProcess_87986700026030
// MI455X (gfx1250) — compile-verified
//
#include <hip/hip_runtime.h>
#include <hip/hip_bf16.h>
#include <cstdint>
#include <cstddef>

// ---------------- problem constants (match reference) ----------------
#define NB     8            // batch
#define NC     80           // classes
#define NH     100
#define NW     152
#define NP     (NH * NW)    // 15200 points
#define PER    (NC * NP)    // 1,216,000 scores per image
#define CHUNK  4096         // elements staged to LDS per block
#define NCHUNK ((PER + CHUNK - 1) / CHUNK)   // 297
#define NBINS  2048         // float-bit-pattern histogram bins (bits >> 20)
#define TOPK   1000         // PRE_NMS_TOP_N
#define SELCAP 4096         // candidate buffer per image (>= TOPK w/ boundary-bin slack)
#define OUTK   100          // POST_NMS_TOP_N
#define NMS_TH 0.6f
#define CLS_TH 0.05f

typedef unsigned int       u32;
typedef unsigned long long u64;

typedef unsigned int v4u __attribute__((ext_vector_type(4)));
typedef int          v4i __attribute__((ext_vector_type(4)));
typedef int          v8i __attribute__((ext_vector_type(8)));

__device__ __forceinline__ float sigf(float x) { return 1.0f / (1.0f + __expf(-x)); }

// ---------------- zero kernels ----------------
__global__ void k_zero_u32(u32* p, int n) {
  int i = blockIdx.x * blockDim.x + threadIdx.x;
  if (i < n) p[i] = 0u;
}
__global__ void k_zero_f32(float* p, int n) {
  int i = blockIdx.x * blockDim.x + threadIdx.x;
  if (i < n) p[i] = 0.0f;
}

// ---------------- LDS staging of one 16KB chunk of box_cls ----------------
// Preferred: Tensor Data Mover DMA (tensor_load_to_lds + s_wait_tensorcnt).
// Fallback 1: global_load_async_to_lds_b128 + s_wait_asynccnt.
// Fallback 2: plain guarded loads.
__device__ __forceinline__ void stage_chunk(const float* __restrict__ gsrc,
                                            float* __restrict__ sdat,
                                            int remain, int tid) {
#if defined(__gfx1250__) && __has_builtin(__builtin_amdgcn_tensor_load_to_lds)
  if (tid < 32) {  // wave 0 issues one DMA descriptor for the whole chunk
    u32 ldsOff = (u32)(size_t)(void*)sdat;            // AS3 offset = low 32 bits
    u64 ga     = (u64)(size_t)(const void*)gsrc;
    u32 dim0   = (remain < CHUNK) ? (u32)remain : (u32)CHUNK;  // OOB reads -> 0
    // D# group 0: count=1 | lds_addr | global_addr[56:0] | type=2
    v4u g0 = { 1u,
               ldsOff,
               (u32)ga,
               (u32)(((ga >> 32) & 0x01FFFFFFu) | 0x80000000u) };
    // D# group 1: data_size=4B; tensor_dim0=dim0; tensor_dim1=1;
    //             tile_dim0=4096; tile_dim1=1; stride0=4096
    v8i g1 = { (int)0x00020000u,
               (int)((dim0 & 0xFFFFu) << 16),
               (int)(((dim0 >> 16) & 0xFFFFu) | (1u << 16)),
               (int)((u32)CHUNK << 16),
               (int)1,
               (int)CHUNK,
               0, 0 };
    v4i g2 = { 0, 0, 0, 0 };
    v4i g3 = { 0, 0, 0, 0 };
    v8i g4 = { 0, 0, 0, 0, 0, 0, 0, 0 };   // extra group (6-arg toolchain), zero-filled
    __builtin_amdgcn_tensor_load_to_lds(g0, g1, g2, g3, g4, 0);
#if __has_builtin(__builtin_amdgcn_s_wait_tensorcnt)
    __builtin_amdgcn_s_wait_tensorcnt(0);
#else
    asm volatile("s_wait_tensorcnt 0" ::: "memory");
#endif
  }
#elif defined(__gfx1250__) && __has_builtin(__builtin_amdgcn_global_load_async_to_lds_b128)
  if (remain >= CHUNK) {
    const char* gb = (const char*)gsrc;
    char* lb = (char*)sdat;
    for (int q = 0; q < 4; ++q) {
      int quad = q * 256 + tid;  // 1024 x 16B quads, coalesced per call
      __builtin_amdgcn_global_load_async_to_lds_b128(
          (__attribute__((address_space(1))) void*)(gb + (size_t)quad * 16),
          (__attribute__((address_space(3))) void*)(lb + (size_t)quad * 16),
          0, 0);
    }
#if __has_builtin(__builtin_amdgcn_s_wait_asynccnt)
    __builtin_amdgcn_s_wait_asynccnt(0);
#else
    asm volatile("s_wait_asynccnt 0" ::: "memory");
#endif
  } else {
    for (int i = 0; i < CHUNK / 256; ++i) {
      int e = tid + i * 256;
      sdat[e] = (e < remain) ? gsrc[e] : 0.0f;
    }
  }
#else
  for (int i = 0; i < CHUNK / 256; ++i) {
    int e = tid + i * 256;
    sdat[e] = (e < remain) ? gsrc[e] : 0.0f;
  }
#endif
}

// ---------------- pass 1: score histogram (per image) ----------------
__global__ __launch_bounds__(256) void k_hist(const float* __restrict__ cls,
                                              const float* __restrict__ ctr,
                                              u32* __restrict__ hist) {
  __shared__ __align__(16) float sdat[CHUNK];
  __shared__ u32 shist[NBINS];
  const int img  = blockIdx.y;
  const int base = blockIdx.x * CHUNK;
  const int tid  = threadIdx.x;
  for (int b = tid; b < NBINS; b += 256) shist[b] = 0u;
  const int remain = PER - base;
  stage_chunk(cls + (size_t)img * PER + base, sdat, remain, tid);
  __syncthreads();
  const float* ctrI = ctr + (size_t)img * NP;
  for (int i = 0; i < CHUNK / 256; ++i) {
    int e = tid + i * 256;
    if (e < remain) {
      int lin = base + e;          // lin = c*NP + p (input layout [C,P])
      float sg = sigf(sdat[e]);
      if (sg > CLS_TH) {
        int p = lin % NP;
        float s = sg * sigf(ctrI[p]);
        if (s > 0.0f) atomicAdd(&shist[__float_as_uint(s) >> 20], 1u);
      }
    }
  }
  __syncthreads();
  u32* h = hist + (size_t)img * NBINS;
  for (int b = tid; b < NBINS; b += 256) {
    u32 v = shist[b];
    if (v) atomicAdd(&h[b], v);
  }
}

// ---------------- pass 2: find top-1000 threshold bin ----------------
__global__ void k_thresh(const u32* __restrict__ hist, u32* __restrict__ thr) {
  int img = threadIdx.x;
  if (img >= NB) return;
  const u32* h = hist + (size_t)img * NBINS;
  u32 acc = 0, B = 0;
  for (int b = NBINS - 1; b >= 0; --b) {
    acc += h[b];
    if (acc >= TOPK) { B = (u32)b; break; }
  }
  thr[img] = B;
}

// ---------------- pass 3: compact candidates >= threshold bin (TDM-staged) ----------------
__global__ __launch_bounds__(256) void k_collect(const float* __restrict__ cls,
                                                 const float* __restrict__ ctr,
                                                 const u32* __restrict__ thr,
                                                 u32* __restrict__ cnt,
                                                 u64* __restrict__ sel) {
  __shared__ __align__(16) float sdat[CHUNK];
  const int img  = blockIdx.y;
  const int base = blockIdx.x * CHUNK;
  const int tid  = threadIdx.x;
  const u32 B = thr[img];
  const int remain = PER - base;
  stage_chunk(cls + (size_t)img * PER + base, sdat, remain, tid);
  __syncthreads();
  const float* ctrI = ctr + (size_t)img * NP;
  u64* selI = sel + (size_t)img * SELCAP;
  for (int i = 0; i < CHUNK / 256; ++i) {
    int e = tid + i * 256;
    if (e < remain) {
      int lin = base + e;
      float sg = sigf(sdat[e]);
      if (sg > CLS_TH) {
        int p = lin % NP, c = lin / NP;
        float s = sg * sigf(ctrI[p]);
        u32 bits = __float_as_uint(s);
        if (s > 0.0f && (bits >> 20) >= B) {
          u32 pos = atomicAdd(&cnt[img], 1u);
          if (pos < SELCAP) {
            u32 idx = (u32)(p * NC + c);          // flat index in [P,C] order
            selI[pos] = ((u64)bits << 32) | (u64)(0xFFFFFFFFu - idx);
          }
        }
      }
    }
  }
}

// ---------------- pass 4: sort + decode + NMS + output (1 block / image) ----------------
__global__ __launch_bounds__(1024) void k_nms(const u64* __restrict__ sel,
                                              const u32* __restrict__ cnt,
                                              const float* __restrict__ locations,
                                              const float* __restrict__ reg,
                                              const float* __restrict__ im_info,
                                              float* __restrict__ out) {
  __shared__ __align__(16) u64 skey[SELCAP];         // 32 KB
  __shared__ float bx1[1024], by1[1024], bx2[1024], by2[1024];
  __shared__ float slbl[1024], sscr[1024];           // 24 KB total box data
  __shared__ int   keep[1024];                       // 4 KB
  const int img = blockIdx.x, tid = threadIdx.x;
  u32 n = cnt[img]; if (n > SELCAP) n = SELCAP;
  const u64* s = sel + (size_t)img * SELCAP;
  for (int i = tid; i < SELCAP; i += 1024) skey[i] = (i < (int)n) ? s[i] : 0ull;
  __syncthreads();
  // bitonic sort, descending (key = score_bits:inv_idx -> exact reference order)
  for (int k = 2; k <= SELCAP; k <<= 1) {
    for (int j = k >> 1; j > 0; j >>= 1) {
      for (int i = tid; i < SELCAP; i += 1024) {
        int l = i ^ j;
        if (l > i) {
          u64 a = skey[i], b = skey[l];
          bool dsc = ((i & k) == 0);
          if (dsc ? (a < b) : (a > b)) { skey[i] = b; skey[l] = a; }
        }
      }
      __syncthreads();
    }
  }
  // decode boxes for the sorted top-1000
  const float h_im = im_info[img * 2 + 0];
  const float w_im = im_info[img * 2 + 1];
  if (tid < TOPK) {
    u64 key = skey[tid];
    float sc = __uint_as_float((u32)(key >> 32));
    float x1 = 0, y1 = 0, x2 = 0, y2 = 0, lb = 0;
    int ok = 0;
    if (sc > 0.0f) {
      u32 idx = 0xFFFFFFFFu - (u32)(key & 0xFFFFFFFFull);
      int p = (int)(idx / NC), c = (int)(idx % NC);
      float px = locations[p * 2 + 0], py = locations[p * 2 + 1];
      const float* r = reg + (size_t)img * 4 * NP;
      float l0 = r[p], t0 = r[NP + p], r0 = r[2 * NP + p], b0 = r[3 * NP + p];
      x1 = fminf(fmaxf(px - l0, 0.0f), w_im - 1.0f);
      y1 = fminf(fmaxf(py - t0, 0.0f), h_im - 1.0f);
      x2 = fminf(fmaxf(px + r0, 0.0f), w_im - 1.0f);
      y2 = fminf(fmaxf(py + b0, 0.0f), h_im - 1.0f);
      lb = (float)(c + 1);
      ok = ((x2 - x1) >= 0.0f) && ((y2 - y1) >= 0.0f);
    }
    bx1[tid] = x1; by1[tid] = y1; bx2[tid] = x2; by2[tid] = y2;
    slbl[tid] = lb; sscr[tid] = ok ? sc : 0.0f; keep[tid] = ok;
  } else {
    sscr[tid] = 0.0f; slbl[tid] = -1.0f; keep[tid] = 0;
  }
  __syncthreads();
  // greedy NMS (score-sorted); class offset trick == label equality check
  for (int i = 0; i < TOPK; ++i) {
    bool alive = (keep[i] != 0);
    if (alive && tid > i && tid < TOPK && slbl[tid] == slbl[i]) {
      float ix1 = fmaxf(bx1[i], bx1[tid]);
      float iy1 = fmaxf(by1[i], by1[tid]);
      float ix2 = fminf(bx2[i], bx2[tid]);
      float iy2 = fminf(by2[i], by2[tid]);
      float inter = fmaxf(ix2 - ix1, 0.0f) * fmaxf(iy2 - iy1, 0.0f);
      float ai = fmaxf(bx2[i] - bx1[i], 0.0f) * fmaxf(by2[i] - by1[i], 0.0f);
      float aj = fmaxf(bx2[tid] - bx1[tid], 0.0f) * fmaxf(by2[tid] - by1[tid], 0.0f);
      float iou = inter / (ai + aj - inter + 1e-9f);
      if (iou > NMS_TH) keep[tid] = 0;
    }
    __syncthreads();
  }
  // emit up to 100 survivors in score order (rest already zeroed)
  if (tid == 0) {
    float* o = out + (size_t)img * OUTK * 6;
    int w = 0;
    for (int j = 0; j < TOPK && w < OUTK; ++j) {
      if (keep[j] && sscr[j] > 0.0f) {
        o[w * 6 + 0] = bx1[j]; o[w * 6 + 1] = by1[j];
        o[w * 6 + 2] = bx2[j]; o[w * 6 + 3] = by2[j];
        o[w * 6 + 4] = slbl[j]; o[w * 6 + 5] = sscr[j];
        ++w;
      }
    }
  }
}

// ---------------- launcher ----------------
extern "C" void kernel_launch(void* const* d_in, const int* in_sizes, int n_in,
                              void* d_out, int out_size, void* d_ws, size_t ws_size,
                              hipStream_t stream) {
  const float* locations = (const float*)d_in[0];  // [P,2]
  const float* box_cls   = (const float*)d_in[1];  // [N,C,H,W]
  const float* box_reg   = (const float*)d_in[2];  // [N,4,H,W]
  const float* ctr       = (const float*)d_in[3];  // [N,1,H,W]
  const float* im_info   = (const float*)d_in[4];  // [N,2]
  float* out = (float*)d_out;

  u32* hist = (u32*)d_ws;                     // NB*NBINS
  u32* cnt  = hist + (size_t)NB * NBINS;      // NB
  u32* thr  = cnt + NB;                       // NB
  size_t selOff = (((size_t)NB * NBINS + 2 * NB) * 4 + 7) & ~(size_t)7;
  u64* sel = (u64*)((char*)d_ws + selOff);    // NB*SELCAP u64

  const int nz = NB * NBINS + 2 * NB;
  k_zero_u32<<<(nz + 255) / 256, 256, 0, stream>>>(hist, nz);
  k_zero_f32<<<(NB * OUTK * 6 + 255) / 256, 256, 0, stream>>>(out, NB * OUTK * 6);

  dim3 gs(NCHUNK, NB);
  k_hist<<<gs, 256, 0, stream>>>(box_cls, ctr, hist);
  k_thresh<<<1, NB, 0, stream>>>(hist, thr);
  k_collect<<<gs, 256, 0, stream>>>(box_cls, ctr, thr, cnt, sel);
  k_nms<<<NB, 1024, 0, stream>>>(sel, cnt, locations, box_reg, im_info, out);
}